// LSTMStateEstimation_54846732370391
// MI455X (gfx1250) — compile-verified
//
#include <hip/hip_runtime.h>

typedef float v2f __attribute__((ext_vector_type(2)));
typedef float v8f __attribute__((ext_vector_type(8)));

#define BATCH 64
#define SEQ   256
#define INPUT 128
#define HID   512
#define NWG   32
#define JSL   16          // hidden units per workgroup: 512/32
#define PHH   (HID + 4)   // padded LDS row stride (516 % 64 == 4 -> conflict-free)

__device__ __forceinline__ float fast_sigmoid(float x) {
    return 1.0f / (1.0f + __expf(-x));
}
__device__ __forceinline__ float fast_tanh(float x) {
    float a = fabsf(x);
    float t = __expf(-2.0f * a);
    float r = (1.0f - t) / (1.0f + t);
    return copysignf(r, x);
}

__global__ void __launch_bounds__(256)
zero_sync_kernel(int* __restrict__ sync) {
    sync[blockIdx.x * 256 + threadIdx.x] = 0;
}

// Persistent cooperative kernel, one per LSTM layer.
// blockIdx.x owns hidden units [j0, j0+16); its 64 gate columns
// (16 each of i,f,g,o) and BOTH weight slices live in LDS for the whole
// sequence. Cell state stays in registers. Steps sync via a device-scope
// arrive/spin barrier.
//
// Dynamic LDS layout (floats):
//   [0 , 4096)                      gbuf[64][64]   gate pre-activations
//   [4096, 4160)                    bias[4][16]
//   [4160, 4160+64*pin)             Wih slice, row stride pin = Kin+4
//   [.. , .. +64*PHH)               Whh slice, row stride 516
__global__ void __launch_bounds__(256, 1)
lstm_layer_kernel(const float* __restrict__ xin, int xbs, int xts, int Kin,
                  const float* __restrict__ Wih, const float* __restrict__ Whh,
                  const float* __restrict__ bih, const float* __restrict__ bhh,
                  float* __restrict__ hseq, int* __restrict__ sync)
{
    extern __shared__ float smem[];
    float* gbuf   = smem;                 // [64][64]
    float* bias   = smem + 4096;          // [4][16]
    const int pin = Kin + 4;
    float* wih    = smem + 4160;          // [64][pin]
    float* whh    = wih + 64 * pin;       // [64][516]

    const int tid   = threadIdx.x;
    const int wave  = tid >> 5;           // 0..7
    const int lane  = tid & 31;
    const int l15   = lane & 15;
    const int khalf = lane >> 4;          // 0: K 0..1, 1: K 2..3
    const int j0    = blockIdx.x * JSL;

    const int mbase  = (wave & 3) << 4;   // batch-tile row base: 0,16,32,48
    const int nblock = wave >> 2;         // 0..1 -> local cols nblock*32..+31

    // ---- one-time: stage bias + weight slices into LDS ----
    if (tid < 64) {
        int gate = tid >> 4, jl = tid & 15;
        int col  = gate * HID + j0 + jl;
        bias[gate * JSL + jl] = bih[col] + bhh[col];
    }
    {
        const int q = Kin >> 2;                       // float4s per row
        for (int idx = tid; idx < 64 * q; idx += 256) {
            int r  = idx / q;
            int kk = (idx - r * q) << 2;
            int col = (r >> 4) * HID + j0 + (r & 15);
            float4 w = *(const float4*)(Wih + (size_t)col * Kin + kk);
            *(float4*)(wih + r * pin + kk) = w;
        }
        const int qh = HID >> 2;
        for (int idx = tid; idx < 64 * qh; idx += 256) {
            int r  = idx / qh;
            int kk = (idx - r * qh) << 2;
            int col = (r >> 4) * HID + j0 + (r & 15);
            float4 w = *(const float4*)(Whh + (size_t)col * HID + kk);
            *(float4*)(whh + r * PHH + kk) = w;
        }
    }

    // Per-lane LDS base pointers for this wave's 2 WMMA B-fragments.
    // local n = nblock*32 + nt*16 + l15
    const float *ihB0, *ihB1, *hhB0, *hhB1;
    {
        int ln0 = nblock * 32 + l15;
        int ln1 = nblock * 32 + 16 + l15;
        ihB0 = wih + ln0 * pin + 2 * khalf;
        ihB1 = wih + ln1 * pin + 2 * khalf;
        hhB0 = whh + ln0 * PHH + 2 * khalf;
        hhB1 = whh + ln1 * PHH + 2 * khalf;
    }

    float creg[4];
    #pragma unroll
    for (int r = 0; r < 4; ++r) creg[r] = 0.0f;

    const float* xrow_base = xin + (size_t)(mbase + l15) * xbs + 2 * khalf;

    __syncthreads();

    for (int t = 0; t < SEQ; ++t) {
        v8f a0 = {}, a1 = {};

        // ---- fused input projection: x_t @ W_ih^T (K = Kin) ----
        const float* xrow = xrow_base + (size_t)t * xts;
        #pragma unroll 8
        for (int k = 0; k < Kin; k += 4) {
            v2f av = *(const v2f*)(xrow + k);
            v2f b0 = *(const v2f*)(ihB0 + k);
            v2f b1 = *(const v2f*)(ihB1 + k);
            a0 = __builtin_amdgcn_wmma_f32_16x16x4_f32(false, av, false, b0, (short)0, a0, false, false);
            a1 = __builtin_amdgcn_wmma_f32_16x16x4_f32(false, av, false, b1, (short)0, a1, false, false);
        }

        // ---- recurrent projection: + h_{t-1} @ W_hh^T (K = 512); h_0 = 0 ----
        if (t > 0) {
            const float* hrow = hseq + (size_t)(t - 1) * BATCH * HID
                              + (size_t)(mbase + l15) * HID + 2 * khalf;
            #pragma unroll 8
            for (int k = 0; k < HID; k += 4) {
                v2f av = *(const v2f*)(hrow + k);
                v2f b0 = *(const v2f*)(hhB0 + k);
                v2f b1 = *(const v2f*)(hhB1 + k);
                a0 = __builtin_amdgcn_wmma_f32_16x16x4_f32(false, av, false, b0, (short)0, a0, false, false);
                a1 = __builtin_amdgcn_wmma_f32_16x16x4_f32(false, av, false, b1, (short)0, a1, false, false);
            }
        }

        // ---- spill D tiles to LDS (D layout: VGPR r -> M = r + 8*khalf) ----
        {
            int ml = mbase + (khalf << 3);
            int nc = nblock * 32 + l15;
            #pragma unroll
            for (int r = 0; r < 8; ++r) {
                gbuf[(ml + r) * 64 + nc +  0] = a0[r];
                gbuf[(ml + r) * 64 + nc + 16] = a1[r];
            }
        }
        __syncthreads();

        // ---- gates + cell/hidden update (c lives in registers) ----
        #pragma unroll
        for (int r = 0; r < 4; ++r) {
            int e  = (r << 8) + tid;          // 0..1023
            int b  = e >> 4;
            int jl = e & 15;
            float iv = fast_sigmoid(gbuf[b * 64 +      jl] + bias[     jl]);
            float fv = fast_sigmoid(gbuf[b * 64 + 16 + jl] + bias[16 + jl]);
            float gv = fast_tanh   (gbuf[b * 64 + 32 + jl] + bias[32 + jl]);
            float ov = fast_sigmoid(gbuf[b * 64 + 48 + jl] + bias[48 + jl]);
            float c  = fv * creg[r] + iv * gv;
            creg[r]  = c;
            hseq[(size_t)t * BATCH * HID + (size_t)b * HID + j0 + jl] = ov * fast_tanh(c);
        }

        // prefetch next timestep's input row while we wait at the barrier
        if (t + 1 < SEQ)
            __builtin_prefetch(xrow_base + (size_t)(t + 1) * xts, 0, 3);

        // ---- device-scope per-step barrier across the 32 workgroups ----
        __threadfence();
        __syncthreads();
        if (tid == 0) {
            __hip_atomic_fetch_add(&sync[t], 1, __ATOMIC_RELEASE, __HIP_MEMORY_SCOPE_AGENT);
            while (__hip_atomic_load(&sync[t], __ATOMIC_ACQUIRE, __HIP_MEMORY_SCOPE_AGENT) < NWG) {
                __builtin_amdgcn_s_sleep(2);
            }
        }
        __syncthreads();
    }
}

__global__ void __launch_bounds__(256)
fc_kernel(const float* __restrict__ h, const float* __restrict__ Wfc,
          const float* __restrict__ bfc, float* __restrict__ out)
{
    int idx = blockIdx.x * 256 + threadIdx.x;   // 0..4095
    int b = idx >> 6, o = idx & 63;
    const float* hr = h + (size_t)b * HID;
    const float* wr = Wfc + (size_t)o * HID;
    float s = 0.0f;
    #pragma unroll 8
    for (int k = 0; k < HID; ++k) s += hr[k] * wr[k];
    out[idx] = s + bfc[o];
}

extern "C" void kernel_launch(void* const* d_in, const int* in_sizes, int n_in,
                              void* d_out, int out_size, void* d_ws, size_t ws_size,
                              hipStream_t stream) {
    (void)in_sizes; (void)n_in; (void)out_size; (void)ws_size;
    const float* x     = (const float*)d_in[0];
    const float* Wih0  = (const float*)d_in[1];
    const float* Whh0  = (const float*)d_in[2];
    const float* bih0  = (const float*)d_in[3];
    const float* bhh0  = (const float*)d_in[4];
    const float* Wih1  = (const float*)d_in[5];
    const float* Whh1  = (const float*)d_in[6];
    const float* bih1  = (const float*)d_in[7];
    const float* bhh1  = (const float*)d_in[8];
    const float* Wfc   = (const float*)d_in[9];
    const float* bfc   = (const float*)d_in[10];

    const size_t HSEQ = (size_t)SEQ * BATCH * HID;   // 8,388,608 floats
    float* h0   = (float*)d_ws;                      // [t][b][512]
    float* h1   = h0 + HSEQ;                         // [t][b][512]
    int*   sync = (int*)(h1 + HSEQ);                 // 512 counters

    zero_sync_kernel<<<2, 256, 0, stream>>>(sync);

    // dynamic LDS: gbuf(4096) + bias(64) + 64*(Kin+4) + 64*516 floats
    size_t lds0 = (size_t)(4096 + 64 + 64 * (INPUT + 4) + 64 * PHH) * sizeof(float);
    size_t lds1 = (size_t)(4096 + 64 + 64 * (HID   + 4) + 64 * PHH) * sizeof(float);

    // Layer 0: x is [b][t][128]  -> batch stride SEQ*INPUT, t stride INPUT
    lstm_layer_kernel<<<NWG, 256, lds0, stream>>>(
        x, SEQ * INPUT, INPUT, INPUT,
        Wih0, Whh0, bih0, bhh0, h0, sync);

    // Layer 1: input is h0 [t][b][512] -> batch stride HID, t stride BATCH*HID
    lstm_layer_kernel<<<NWG, 256, lds1, stream>>>(
        h0, HID, BATCH * HID, HID,
        Wih1, Whh1, bih1, bhh1, h1, sync + SEQ);

    // FC on last timestep of layer 1
    fc_kernel<<<16, 256, 0, stream>>>(h1 + (size_t)(SEQ - 1) * BATCH * HID,
                                      Wfc, bfc, (float*)d_out);
}